// PsRoIAlign_19421842113057
// MI455X (gfx1250) — compile-verified
//
#include <hip/hip_runtime.h>
#include <hip/hip_bf16.h>
#include <math.h>

#define OUT_H 7
#define OUT_W 7
#define NBIN  (OUT_H * OUT_W)      // 49
#define GS    2                    // SAMPLE_NUM
#define SPATIAL_SCALE 0.125f
#define NAX   (OUT_H * GS)         // 14 axis sample entries per dimension

// mmcv-style per-axis bilinear terms (matches reference _axis_interp):
//   valid = (c >= -1) && (c <= size); cc = max(c,0); i0 = floor(cc)
//   if i0 >= size-1: ilow=ihigh=size-1, l=0   else ilow=i0, ihigh=i0+1, l=cc-i0
// Weights returned PRE-SCALED by the validity mask (valid distributes over the
// bilinear sum as exactly one y-weight * one x-weight per term).
__device__ __forceinline__ void axis_interp(float c, int size,
                                            int& ilow, int& ihigh,
                                            float& l, float& h) {
    const float valid = (c >= -1.0f && c <= (float)size) ? 1.0f : 0.0f;
    float cc = fmaxf(c, 0.0f);
    int i0 = (int)floorf(cc);
    if (i0 >= size - 1) {
        ilow = size - 1; ihigh = size - 1; l = 0.0f;
    } else {
        ilow = i0; ihigh = i0 + 1; l = cc - (float)i0;
    }
    h = (1.0f - l) * valid;
    l = l * valid;
}

__global__ void __launch_bounds__(256)
psroi_align_kernel(const float* __restrict__ feat,   // [N, C, H, W]
                   const float* __restrict__ rois,   // [R, 5]
                   float* __restrict__ out,          // [R, c_out, 7, 7]
                   int C, int H, int W, int c_out) {
    const int roi = blockIdx.x;
    const int tid = threadIdx.x;

    __shared__ float s_roi[8];                 // 5 dwords used (async-to-LDS target)
    __shared__ int   s_y0[NAX], s_y1[NAX];     // pre-multiplied by W
    __shared__ int   s_x0[NAX], s_x1[NAX];
    __shared__ float s_ly[NAX], s_hy[NAX];     // pre-scaled by y-validity
    __shared__ float s_lx[NAX], s_hx[NAX];     // pre-scaled by x-validity

    // ---- CDNA5 async gather of the RoI descriptor into LDS (ASYNCcnt path) ----
    if (tid < 5) {
        unsigned int lds_addr =
            (unsigned int)(unsigned long long)(&s_roi[tid]);   // low 32b of generic ptr == LDS offset
        const float* gaddr = rois + (size_t)roi * 5 + tid;
        asm volatile("global_load_async_to_lds_b32 %0, %1, off"
                     :: "v"(lds_addr), "v"(gaddr)
                     : "memory");
    }
    asm volatile("s_wait_asynccnt 0" ::: "memory");
    __syncthreads();

    // ---- per-RoI axis tables, computed once, amortized over 490 outputs ----
    if (tid < 2 * NAX) {
        const float sx = fmaf(s_roi[1], SPATIAL_SCALE, -0.5f);
        const float sy = fmaf(s_roi[2], SPATIAL_SCALE, -0.5f);
        const float ex = fmaf(s_roi[3], SPATIAL_SCALE, -0.5f);
        const float ey = fmaf(s_roi[4], SPATIAL_SCALE, -0.5f);
        const float bw = fmaxf(ex - sx, 0.1f) / (float)OUT_W;
        const float bh = fmaxf(ey - sy, 0.1f) / (float)OUT_H;

        const int axis = (tid < NAX) ? 0 : 1;       // 0 = y, 1 = x
        const int i    = (tid < NAX) ? tid : tid - NAX;
        const int p    = i >> 1;                    // bin index along axis
        const int s    = i & 1;                     // sub-sample index
        const float off = ((float)s + 0.5f) / (float)GS;

        int ilow, ihigh; float l, h;
        if (axis == 0) {
            float y = fmaf((float)p + off, bh, sy);
            axis_interp(y, H, ilow, ihigh, l, h);
            s_y0[i] = ilow * W; s_y1[i] = ihigh * W;   // row element-offsets ready
            s_ly[i] = l; s_hy[i] = h;
        } else {
            float x = fmaf((float)p + off, bw, sx);
            axis_interp(x, W, ilow, ihigh, l, h);
            s_x0[i] = ilow; s_x1[i] = ihigh;
            s_lx[i] = l; s_hx[i] = h;
        }
    }
    __syncthreads();

    const int b   = (int)s_roi[0];
    const int HW  = H * W;
    const int tot = c_out * NBIN;                  // 490 outputs per RoI
    float* out_roi = out + (size_t)roi * tot;

    for (int o = tid; o < tot; o += blockDim.x) {
        const int c   = o / NBIN;
        const int bin = o - c * NBIN;              // c_in = c*49 + bin
        const int ph  = bin / OUT_W;
        const int pw  = bin - ph * OUT_W;

        const float* __restrict__ base = feat + (size_t)(b * C + c * NBIN + bin) * HW;

        float acc = 0.0f;
        #pragma unroll
        for (int syi = 0; syi < GS; ++syi) {
            const int yi = ph * GS + syi;
            const float ly = s_ly[yi], hy = s_hy[yi];
            const float* __restrict__ r0 = base + s_y0[yi];
            const float* __restrict__ r1 = base + s_y1[yi];
            #pragma unroll
            for (int sxi = 0; sxi < GS; ++sxi) {
                const int xi = pw * GS + sxi;
                const int x0 = s_x0[xi], x1 = s_x1[xi];
                const float lx = s_lx[xi], hx = s_hx[xi];
                // 4 independent gathers -> overlapped under LOADcnt
                const float v00 = r0[x0];
                const float v01 = r0[x1];
                const float v10 = r1[x0];
                const float v11 = r1[x1];
                acc += hy * hx * v00 + hy * lx * v01
                     + ly * hx * v10 + ly * lx * v11;   // validity pre-folded
            }
        }
        __builtin_nontemporal_store(acc * 0.25f, &out_roi[o]);   // / (g*g); NT: never re-read
    }
}

extern "C" void kernel_launch(void* const* d_in, const int* in_sizes, int n_in,
                              void* d_out, int out_size, void* d_ws, size_t ws_size,
                              hipStream_t stream) {
    const float* feat = (const float*)d_in[0];
    const float* rois = (const float*)d_in[1];
    float* out = (float*)d_out;

    const int R = in_sizes[1] / 5;                 // 1024
    const int N = 4, H = 168, W = 168;             // per reference setup_inputs
    const int C = in_sizes[0] / (N * H * W);       // 490
    const int c_out = C / NBIN;                    // 10

    dim3 grid(R), block(256);
    psroi_align_kernel<<<grid, block, 0, stream>>>(feat, rois, out, C, H, W, c_out);
}